// BQFeatureInjector_30648886624904
// MI455X (gfx1250) — compile-verified
//
#include <hip/hip_runtime.h>
#include <stdint.h>

#define HID 256
#define BQH 128
#define CAP 2048   // compacted in-radius candidate capacity (exact fallback if exceeded)

typedef __attribute__((ext_vector_type(2))) float v2f;
typedef __attribute__((ext_vector_type(8))) float v8f;

// ---------------------------------------------------------------------------
// Kernel 1: out = concat(geo_tokens, surf_tokens, vol_tokens)
// ---------------------------------------------------------------------------
__global__ __launch_bounds__(256)
void copy_tokens(const float* __restrict__ g, const float* __restrict__ s,
                 const float* __restrict__ v, float* __restrict__ out)
{
    size_t i = (size_t)blockIdx.x * 256 + threadIdx.x;
    const size_t NGf = (size_t)4096 * HID;
    const size_t NSf = (size_t)4096 * HID;
    const size_t NVf = (size_t)8192 * HID;
    if (i < NGf)                  out[i] = g[i];
    else if (i < NGf + NSf)       out[i] = s[i - NGf];
    else if (i < NGf + NSf + NVf) out[i] = v[i - NGf - NSf];
}

// ---------------------------------------------------------------------------
// Kernel 2: per-query ball-query features. One block (256 threads) per query.
//  - d^2 for all Np points into LDS; compact indices of d^2 <= r1^2 candidates
//  - exact nearest-32 among candidates via 32 passes of (d2,idx) u64-min,
//    wave32 __shfl_down reduction + tiny cross-wave LDS reduce
//  - per-scale masked max-pool of relu(rel @ W1 + b1) -> 256-dim feature row
// Correctness note: the reference's contributing set {top-k overall} ∩
// {d^2<=r^2} equals the nearest min(k, cnt) in-radius points, which is
// exactly what this selection produces (ties broken by lower index, matching
// jax.lax.top_k).
// ---------------------------------------------------------------------------
__global__ __launch_bounds__(256)
void bq_features(const float* __restrict__ Q, const float* __restrict__ P,
                 int Np,
                 const float* __restrict__ W1s, const float* __restrict__ b1s,
                 float r0sq, float r1sq,
                 float* __restrict__ F)
{
    __shared__ float s_d2[8192];
    __shared__ int   s_cand[CAP];
    __shared__ int   s_cnt;
    __shared__ unsigned long long s_red[8];
    __shared__ unsigned long long s_prevk;
    __shared__ float s_rel[32 * 3];
    __shared__ float s_seld2[32];
    __shared__ float s_w1[2 * 3 * BQH];
    __shared__ float s_b1[2 * BQH];
    __shared__ float s_q[3];

    const int tid  = threadIdx.x;
    const int lane = tid & 31;
    const int wv   = tid >> 5;
    const int q    = blockIdx.x;

    if (tid < 3) s_q[tid] = Q[(size_t)q * 3 + tid];
    for (int i = tid; i < 2 * 3 * BQH; i += 256) s_w1[i] = W1s[i];
    for (int i = tid; i < 2 * BQH;     i += 256) s_b1[i] = b1s[i];
    if (tid == 0) { s_prevk = 0ull; s_cnt = 0; }
    __syncthreads();

    const float qx = s_q[0], qy = s_q[1], qz = s_q[2];

    // d^2 for all points + compaction of in-radius(r1) candidates.
    for (int i = tid; i < Np; i += 256) {
        float dx = P[i * 3 + 0] - qx;
        float dy = P[i * 3 + 1] - qy;
        float dz = P[i * 3 + 2] - qz;
        float d2 = dx * dx + dy * dy + dz * dz;
        s_d2[i] = d2;
        if (d2 <= r1sq) {
            int pos = atomicAdd(&s_cnt, 1);
            if (pos < CAP) s_cand[pos] = i;
        }
    }
    __syncthreads();

    const int  cnt        = s_cnt;
    const bool useCompact = (cnt <= CAP);
    const int  M          = useCompact ? cnt : Np;

    // 32 selection passes: strictly increasing (d2bits<<32 | idx+1) keys.
    for (int j = 0; j < 32; ++j) {
        const unsigned long long prev = s_prevk;
        unsigned long long best = ~0ull;
        for (int t = tid; t < M; t += 256) {
            const int i = useCompact ? s_cand[t] : t;
            unsigned int db = __float_as_uint(s_d2[i]);
            unsigned long long key =
                ((unsigned long long)db << 32) | (unsigned int)(i + 1);
            if (key > prev && key < best) best = key;
        }
        // wave32-internal u64 min (no barriers)
#pragma unroll
        for (int off = 16; off > 0; off >>= 1) {
            unsigned long long o = __shfl_down(best, off, 32);
            if (o < best) best = o;
        }
        if (lane == 0) s_red[wv] = best;
        __syncthreads();
        if (tid == 0) {
            unsigned long long sel = s_red[0];
#pragma unroll
            for (int w = 1; w < 8; ++w)
                if (s_red[w] < sel) sel = s_red[w];
            s_prevk = sel;
            if (sel != ~0ull) {
                int idx = (int)(sel & 0xffffffffu) - 1;
                s_seld2[j]       = __uint_as_float((unsigned int)(sel >> 32));
                s_rel[j * 3 + 0] = P[(size_t)idx * 3 + 0] - qx;
                s_rel[j * 3 + 1] = P[(size_t)idx * 3 + 1] - qy;
                s_rel[j * 3 + 2] = P[(size_t)idx * 3 + 2] - qz;
            } else {
                s_seld2[j]       = 1e30f;   // fails every radius test
                s_rel[j * 3 + 0] = 0.0f;
                s_rel[j * 3 + 1] = 0.0f;
                s_rel[j * 3 + 2] = 0.0f;
            }
        }
        __syncthreads();
    }

    // Masked max-pool of relu(rel . W1[:,h] + b1[h]) over k nearest in radius.
    // relu >= 0 and "no valid neighbor" -> 0, so init best = 0 is exact.
    const int   sc = tid >> 7;          // scale 0 (k=16,r0) / 1 (k=32,r1)
    const int   h  = tid & (BQH - 1);
    const int   kk = sc ? 32 : 16;
    const float r2 = sc ? r1sq : r0sq;
    const float w0 = s_w1[(sc * 3 + 0) * BQH + h];
    const float w1 = s_w1[(sc * 3 + 1) * BQH + h];
    const float w2 = s_w1[(sc * 3 + 2) * BQH + h];
    const float bb = s_b1[sc * BQH + h];
    float best = 0.0f;
    for (int j = 0; j < kk; ++j) {
        if (s_seld2[j] <= r2) {
            float val = fmaf(s_rel[j * 3 + 0], w0,
                        fmaf(s_rel[j * 3 + 1], w1,
                        fmaf(s_rel[j * 3 + 2], w2, bb)));
            val = val > 0.0f ? val : 0.0f;
            if (val > best) best = val;
        }
    }
    F[(size_t)q * HID + sc * BQH + h] = best;
}

// ---------------------------------------------------------------------------
// Kernel 3: Out[m0+..][n0+..] += F @ W2s + b2s using V_WMMA_F32_16X16X4_F32.
// One wave per 16x16 tile, 4 waves per block. K = 256 in steps of 4.
// A(16x4) f32 layout: lanes 0-15 -> M=lane, VGPR{0,1}=K{0,1};
//                     lanes 16-31 -> M=lane-16, VGPR{0,1}=K{2,3}.
// C/D(16x16): VGPR v, lanes 0-15 -> M=v, N=lane; lanes 16-31 -> M=v+8.
// ---------------------------------------------------------------------------
__global__ __launch_bounds__(128)
void bq_gemm_wmma(const float* __restrict__ F, const float* __restrict__ W2s,
                  const float* __restrict__ b2s, float* __restrict__ Out,
                  int Nq)
{
    const int lane = threadIdx.x & 31;
    const int wave = threadIdx.x >> 5;
    const int tile = blockIdx.x * 4 + wave;
    const int ntn  = HID / 16;             // 16 tiles along N
    const int m0   = (tile / ntn) * 16;    // query-row base
    const int n0   = (tile % ntn) * 16;    // output-col base
    if (m0 >= Nq) return;

    const int r  = lane & 15;   // A: M row / B: N col / D: N col
    const int kh = lane >> 4;   // which K-half of the 4-wide step

    v8f acc = {};
    const float* fa = F + (size_t)(m0 + r) * HID + 2 * kh;
#pragma unroll 4
    for (int k0 = 0; k0 < HID; k0 += 4) {
        v2f a, b;
        a.x = fa[k0 + 0];
        a.y = fa[k0 + 1];
        b.x = W2s[(size_t)(k0 + 2 * kh + 0) * HID + n0 + r];
        b.y = W2s[(size_t)(k0 + 2 * kh + 1) * HID + n0 + r];
        acc = __builtin_amdgcn_wmma_f32_16x16x4_f32(
            false, a, false, b, (short)0, acc, false, false);
    }

    const float bv = b2s[n0 + r];
#pragma unroll
    for (int v = 0; v < 8; ++v) {
        const int mrow = v + 8 * kh;
        float* o = Out + (size_t)(m0 + mrow) * HID + n0 + r;
        *o = *o + acc[v] + bv;
    }
}

// ---------------------------------------------------------------------------
extern "C" void kernel_launch(void* const* d_in, const int* in_sizes, int n_in,
                              void* d_out, int out_size, void* d_ws, size_t ws_size,
                              hipStream_t stream)
{
    (void)in_sizes; (void)n_in; (void)out_size; (void)ws_size;

    const float* gpts = (const float*)d_in[0];
    const float* spts = (const float*)d_in[1];
    const float* vpts = (const float*)d_in[2];
    const float* gtok = (const float*)d_in[3];
    const float* stok = (const float*)d_in[4];
    const float* vtok = (const float*)d_in[5];
    const float* W1   = (const float*)d_in[6];
    const float* b1   = (const float*)d_in[7];
    const float* W2   = (const float*)d_in[8];
    const float* b2   = (const float*)d_in[9];
    float* out = (float*)d_out;

    const int NG = 4096, NS = 4096, NV = 8192;
    float* gout = out;
    float* sout = out + (size_t)NG * HID;
    float* vout = out + (size_t)(NG + NS) * HID;
    float* F    = (float*)d_ws;   // NV * HID floats = 8 MB scratch

    {
        size_t total = (size_t)(NG + NS + NV) * HID;
        copy_tokens<<<(unsigned)(total / 256), 256, 0, stream>>>(gtok, stok, vtok, out);
    }

    const float R0 = 0.1f, R1 = 0.2f;
    struct StackDesc { int s; const float* q; int nq; const float* p; int np; float* o; };
    const StackDesc st[7] = {
        {0, gpts, NG, gpts, NG, gout},   // geo self
        {3, gpts, NG, spts, NS, gout},   // geo <- surf
        {1, spts, NS, spts, NS, sout},   // surf self
        {5, spts, NS, gpts, NG, sout},   // surf <- geo
        {4, gpts, NG, vpts, NV, gout},   // geo <- vol
        {2, vpts, NV, vpts, NV, vout},   // vol self
        {6, vpts, NV, gpts, NG, vout},   // vol <- geo
    };

    for (int i = 0; i < 7; ++i) {
        const StackDesc& d = st[i];
        const float* W1s = W1 + (size_t)d.s * 2 * 3 * BQH;
        const float* b1s = b1 + (size_t)d.s * 2 * BQH;
        const float* W2s = W2 + (size_t)d.s * 2 * BQH * HID;
        const float* b2s = b2 + (size_t)d.s * HID;

        bq_features<<<d.nq, 256, 0, stream>>>(d.q, d.p, d.np, W1s, b1s,
                                              R0 * R0, R1 * R1, F);

        const int tiles = (d.nq / 16) * (HID / 16);
        bq_gemm_wmma<<<tiles / 4, 128, 0, stream>>>(F, W2s, b2s, d.o, d.nq);
    }
}